// MyModel_67516885893171
// MI455X (gfx1250) — compile-verified
//
#include <hip/hip_runtime.h>

typedef __attribute__((ext_vector_type(2))) float v2f;
typedef __attribute__((ext_vector_type(8))) float v8f;

#define NCOMP 5
#define BISECT_ITERS 60

__device__ __forceinline__ float fexp2(float x) { return __builtin_amdgcn_exp2f(x); }
__device__ __forceinline__ float flog2(float x) { return __builtin_amdgcn_logf(x); }   // log2
__device__ __forceinline__ float frcp(float x)  { return __builtin_amdgcn_rcpf(x); }

// Sign test of g(lam) = KL(p(lam)||r) - eps with dd_i = log2(q_i) - max (<= 0).
// g = ln2*(T/S - log2 S) + (ln5 - eps), S > 0, so
// g > 0  <=>  T + S*(CC2 - log2 S) > 0,  CC2 = (ln5 - eps)/ln2.
__device__ __forceinline__ bool g_pos(const float* dd, float lam) {
    const float CC2 = 2.17765859079846660f;            // log2(5) - 0.1/ln2
    float t = frcp(1.0f + lam);
    float S = 0.0f, T = 0.0f;
    #pragma unroll
    for (int i = 0; i < NCOMP; ++i) {
        float z = t * dd[i];                           // max pre-shifted: z <= 0
        float e = fexp2(z);
        S += e;
        T = fmaf(e, z, T);
    }
    return fmaf(S, CC2 - flog2(S), T) > 0.0f;
}

__device__ __forceinline__ void solve_row(const float* q, float* p) {
    // dd_i = log2(q_i) - max_j log2(q_j); uniform shifts cancel in softmax & KL
    float dd[NCOMP], dmax = -3.0e38f;
    #pragma unroll
    for (int i = 0; i < NCOMP; ++i) { dd[i] = flog2(q[i]); dmax = fmaxf(dmax, dd[i]); }
    #pragma unroll
    for (int i = 0; i < NCOMP; ++i) dd[i] -= dmax;

    bool infeasible0 = g_pos(dd, 0.0f);                // g(0) > 0 ?
    float lo = 0.0f, hi = 1.0e6f;
    #pragma unroll 1
    for (int it = 0; it < BISECT_ITERS; ++it) {        // branchless bisection
        float mid = 0.5f * (lo + hi);
        bool pos = g_pos(dd, mid);
        lo = pos ? mid : lo;
        hi = pos ? hi : mid;
    }
    float lam = infeasible0 ? 0.5f * (lo + hi) : 0.0f; // exact 0 if feasible
    float t = frcp(1.0f + lam);                        // rcp(1.0) == 1.0 exactly
    float e[NCOMP], S = 0.0f;
    #pragma unroll
    for (int i = 0; i < NCOMP; ++i) { e[i] = fexp2(t * dd[i]); S += e[i]; }
    float inv = 1.0f / S;
    #pragma unroll
    for (int i = 0; i < NCOMP; ++i) p[i] = e[i] * inv;
}

// One wave = 32 rows = two 16-row WMMA tiles. q = x*W^T + b via
// V_WMMA_F32_16X16X4_F32 (two issues: K=0..3 and K=4 in slot 0).
__global__ __launch_bounds__(256) void klproj_wmma_kernel(
    const float* __restrict__ x, const float* __restrict__ W,
    const float* __restrict__ bvec, float* __restrict__ out, int B)
{
    __shared__ float lds[8 * 160];                    // per-wave 32 rows x 5 floats
    const int lane  = threadIdx.x & 31;
    const int wIB   = threadIdx.x >> 5;
    const int gwave = blockIdx.x * 8 + wIB;
    const int nwave = gridDim.x * 8;
    float* wls = &lds[wIB * 160];

    const int n    = lane & 15;                       // output column (N)
    const int hi   = lane >> 4;                       // lane half
    const int koff = hi << 1;                         // K slots 0,1 / 2,3

    // B-matrix (W^T, zero-padded 5->16 cols) + bias-broadcast C, built once.
    float b1x = 0.f, b1y = 0.f, b2x = 0.f, cbias = 0.f;
    if (n < NCOMP) {
        b1x = W[n * NCOMP + koff];                    // W^T[k=koff][n]
        b1y = W[n * NCOMP + koff + 1];                // W^T[k=koff+1][n]
        if (!hi) b2x = W[n * NCOMP + 4];              // k=4 placed in K-slot 0
        cbias = bvec[n];
    }
    v2f Bf1; Bf1.x = b1x; Bf1.y = b1y;
    v2f Bf2; Bf2.x = b2x; Bf2.y = 0.f;

    const int nChunks = B >> 5;                       // full 32-row chunks
    for (int c = gwave; c < nChunks; c += nwave) {
        const long base5 = (long)c * 160;
        // Coalesced stage-in of 32 rows x 5 floats (5 x 128B transactions)
        #pragma unroll
        for (int j = 0; j < 5; ++j)
            wls[j * 32 + lane] = x[base5 + j * 32 + lane];
        asm volatile("s_wait_dscnt 0" ::: "memory");

        #pragma unroll
        for (int t = 0; t < 2; ++t) {
            const int mrow = t * 16 + n;              // A row for this lane
            v2f Af1, Af2;
            Af1.x = wls[mrow * 5 + koff];
            Af1.y = wls[mrow * 5 + koff + 1];
            float a4 = wls[mrow * 5 + 4];
            Af2.x = hi ? 0.f : a4;                    // K-slot 0 carries k=4
            Af2.y = 0.f;
            v8f acc;
            #pragma unroll
            for (int i = 0; i < 8; ++i) acc[i] = cbias;
            acc = __builtin_amdgcn_wmma_f32_16x16x4_f32(
                      false, Af1, false, Bf1, (short)0, acc, false, false);
            acc = __builtin_amdgcn_wmma_f32_16x16x4_f32(
                      false, Af2, false, Bf2, (short)0, acc, false, false);
            // Scatter q tile to LDS (D layout: VGPR v -> M = v + hi*8, N = n)
            if (n < NCOMP) {
                #pragma unroll
                for (int v = 0; v < 8; ++v) {
                    int m = t * 16 + v + (hi << 3);
                    wls[m * 5 + n] = acc[v];
                }
            }
        }
        asm volatile("s_wait_dscnt 0" ::: "memory");

        // One row per lane: 60-step bisection (stride 5 => conflict-free LDS)
        float q[NCOMP], p[NCOMP];
        #pragma unroll
        for (int i = 0; i < NCOMP; ++i) q[i] = wls[lane * 5 + i];
        solve_row(q, p);

        // Stage-out through LDS for coalesced stores
        #pragma unroll
        for (int i = 0; i < NCOMP; ++i) wls[lane * 5 + i] = p[i];
        asm volatile("s_wait_dscnt 0" ::: "memory");
        #pragma unroll
        for (int j = 0; j < 5; ++j)
            out[base5 + j * 32 + lane] = wls[j * 32 + lane];
    }

    // Scalar tail for B % 32 rows (empty for B = 2^21, kept for generality)
    const int tailStart = nChunks << 5;
    for (int row = tailStart + (int)(blockIdx.x * blockDim.x + threadIdx.x);
         row < B; row += (int)(gridDim.x * blockDim.x)) {
        float q[NCOMP], p[NCOMP];
        #pragma unroll
        for (int i = 0; i < NCOMP; ++i) {
            float acc = bvec[i];
            #pragma unroll
            for (int k = 0; k < NCOMP; ++k)
                acc += x[(long)row * NCOMP + k] * W[i * NCOMP + k];
            q[i] = acc;
        }
        solve_row(q, p);
        #pragma unroll
        for (int i = 0; i < NCOMP; ++i) out[(long)row * NCOMP + i] = p[i];
    }
}

extern "C" void kernel_launch(void* const* d_in, const int* in_sizes, int n_in,
                              void* d_out, int out_size, void* d_ws, size_t ws_size,
                              hipStream_t stream) {
    const float* x = (const float*)d_in[0];
    const float* W = (const float*)d_in[1];
    const float* b = (const float*)d_in[2];
    float* out = (float*)d_out;
    const int B = in_sizes[0] / NCOMP;

    const int nChunks = B >> 5;
    int blocks = (nChunks + 7) / 8;                   // 8 waves per 256-thread block
    if (blocks > 4096) blocks = 4096;
    if (blocks < 1) blocks = 1;
    klproj_wmma_kernel<<<blocks, 256, 0, stream>>>(x, W, b, out, B);
}